// GATAttentionAggregator_27736898798016
// MI455X (gfx1250) — compile-verified
//
#include <hip/hip_runtime.h>
#include <cstdint>
#include <cstddef>

#define NUM_H 4
#define DIM 128
#define NEG_SLOPE 0.2f

typedef __attribute__((ext_vector_type(2))) float v2f;
typedef __attribute__((ext_vector_type(8))) float v8f;

// ---- monotonic float <-> uint mapping (for integer atomicMax on floats) ----
__device__ __forceinline__ unsigned f2ord(float f) {
  unsigned u = __float_as_uint(f);
  return (u & 0x80000000u) ? ~u : (u | 0x80000000u);
}
__device__ __forceinline__ float ord2f(unsigned o) {
  unsigned u = (o & 0x80000000u) ? (o & 0x7fffffffu) : ~o;
  return __uint_as_float(u);
}

// ---------------- K0: zero workspace prefix (agg + maxord + denom) ----------
__global__ void k_zero(uint4* __restrict__ p, long n4) {
  long i = (long)blockIdx.x * blockDim.x + threadIdx.x;
  long stride = (long)gridDim.x * blockDim.x;
  uint4 z = {0u, 0u, 0u, 0u};
  for (; i < n4; i += stride) p[i] = z;
}

// ---------------- K1: scores[e][h] = LeakyReLU(msg[e] . A[h]) ---------------
// wave-per-edge: lane l loads msg[e][4l..4l+3] (512B contiguous per wave),
// 4 dot-products reduced with wave32 shuffles.
__global__ void k_scores(const float* __restrict__ msg, const float* __restrict__ A,
                         float* __restrict__ scores, int E) {
  __shared__ float sA[NUM_H * DIM];
  for (int i = threadIdx.x; i < NUM_H * DIM; i += blockDim.x) sA[i] = A[i];
  __syncthreads();
  const int lane = threadIdx.x & 31;
  const int wid  = blockIdx.x * (blockDim.x >> 5) + (threadIdx.x >> 5);
  const int nw   = gridDim.x * (blockDim.x >> 5);
  const float4 a0 = ((const float4*)(sA + 0 * DIM))[lane];
  const float4 a1 = ((const float4*)(sA + 1 * DIM))[lane];
  const float4 a2 = ((const float4*)(sA + 2 * DIM))[lane];
  const float4 a3 = ((const float4*)(sA + 3 * DIM))[lane];
  for (int e = wid; e < E; e += nw) {
    const float4 m = ((const float4*)(msg + (size_t)e * DIM))[lane];
    float s0 = m.x * a0.x + m.y * a0.y + m.z * a0.z + m.w * a0.w;
    float s1 = m.x * a1.x + m.y * a1.y + m.z * a1.z + m.w * a1.w;
    float s2 = m.x * a2.x + m.y * a2.y + m.z * a2.z + m.w * a2.w;
    float s3 = m.x * a3.x + m.y * a3.y + m.z * a3.z + m.w * a3.w;
#pragma unroll
    for (int off = 16; off > 0; off >>= 1) {
      s0 += __shfl_xor(s0, off, 32);
      s1 += __shfl_xor(s1, off, 32);
      s2 += __shfl_xor(s2, off, 32);
      s3 += __shfl_xor(s3, off, 32);
    }
    if (lane == 0) {
      float4 r;
      r.x = (s0 >= 0.f) ? s0 : NEG_SLOPE * s0;
      r.y = (s1 >= 0.f) ? s1 : NEG_SLOPE * s1;
      r.z = (s2 >= 0.f) ? s2 : NEG_SLOPE * s2;
      r.w = (s3 >= 0.f) ? s3 : NEG_SLOPE * s3;
      *(float4*)(scores + (size_t)e * NUM_H) = r;
    }
  }
}

// ---------------- K2: segment max (atomicMax on ordered-uint) ---------------
__global__ void k_segmax(const float* __restrict__ scores, const int* __restrict__ index,
                         unsigned* __restrict__ maxord, long EH) {
  long i = (long)blockIdx.x * blockDim.x + threadIdx.x;
  long stride = (long)gridDim.x * blockDim.x;
  for (; i < EH; i += stride) {
    long e = i >> 2;
    int  h = (int)(i & 3);
    atomicMax(&maxord[(size_t)index[e] * NUM_H + h], f2ord(scores[i]));
  }
}

// ---------------- K3: e = exp(s - max); denom += e ---------------------------
__global__ void k_expsum(float* __restrict__ scores, const int* __restrict__ index,
                         const unsigned* __restrict__ maxord, float* __restrict__ denom,
                         long EH) {
  long i = (long)blockIdx.x * blockDim.x + threadIdx.x;
  long stride = (long)gridDim.x * blockDim.x;
  for (; i < EH; i += stride) {
    long e = i >> 2;
    int  h = (int)(i & 3);
    size_t nh = (size_t)index[e] * NUM_H + h;
    float v = __expf(scores[i] - ord2f(maxord[nh]));
    scores[i] = v;                 // overwrite score with exp value
    atomicAdd(&denom[nh], v);
  }
}

// ---------------- K4: agg[n][h][:] += alpha[e][h] * msg[e][:] ---------------
// wave-per-edge, coalesced msg read, 16 f32 atomic adds per lane per edge
// (512 contiguous floats per edge -> well-coalesced atomic traffic to L2).
__global__ void k_agg(const float* __restrict__ msg, const int* __restrict__ index,
                      const float* __restrict__ scores, const float* __restrict__ denom,
                      float* __restrict__ agg, int E) {
  const int lane = threadIdx.x & 31;
  const int wid  = blockIdx.x * (blockDim.x >> 5) + (threadIdx.x >> 5);
  const int nw   = gridDim.x * (blockDim.x >> 5);
  for (int e = wid; e < E; e += nw) {
    const int idx = index[e];
    const float4 ev = *(const float4*)(scores + (size_t)e * NUM_H);
    const float4 dn = *(const float4*)(denom + (size_t)idx * NUM_H);
    float al[NUM_H];
    al[0] = ev.x / dn.x; al[1] = ev.y / dn.y; al[2] = ev.z / dn.z; al[3] = ev.w / dn.w;
    const float4 m = ((const float4*)(msg + (size_t)e * DIM))[lane];
    float* base = agg + (size_t)idx * NUM_H * DIM + lane * 4;
#pragma unroll
    for (int h = 0; h < NUM_H; ++h) {
      const float a = al[h];
      float* p = base + h * DIM;
      atomicAdd(p + 0, a * m.x);
      atomicAdd(p + 1, a * m.y);
      atomicAdd(p + 2, a * m.z);
      atomicAdd(p + 3, a * m.w);
    }
  }
}

// ---------------- K5: out = 0.25 * sum_h agg[:, h, :] @ W[h]^T  (WMMA f32) --
// One workgroup per 16-node tile. 8 waves each own a 16-col tile of the
// 128-wide output. agg tile staged in padded LDS (stride 132 -> no 16-way
// bank conflicts on A-fragment reads). 4 heads x 32 K-steps of
// V_WMMA_F32_16X16X4_F32 accumulate in a single v8f.
//
// A-frag layout (ISA 7.12.2, 32-bit 16x4): v0: lanes0-15 K=k+0, lanes16-31 K=k+2
//                                          v1: lanes0-15 K=k+1, lanes16-31 K=k+3
// B-frag mirrors with N=lane%16 (B[k][n] = W[h][n][k]).
// C/D layout: VGPR r -> M=r (lanes0-15) / M=8+r (lanes16-31), N=lane%16.
__global__ void __launch_bounds__(256) k_out(const float* __restrict__ agg,
                                             const float* __restrict__ W,
                                             float* __restrict__ out, int N) {
  __shared__ float tile[16 * 132];
  const int node0 = blockIdx.x * 16;
  const int tid  = threadIdx.x;
  const int wave = tid >> 5;
  const int lane = tid & 31;
  const int lh   = lane & 15;
  const int khi  = (lane >> 4) << 1;   // 0 for lanes 0-15, 2 for lanes 16-31
  const int c0   = wave * 16;
  v8f acc = {0.f, 0.f, 0.f, 0.f, 0.f, 0.f, 0.f, 0.f};

  for (int h = 0; h < NUM_H; ++h) {
    __syncthreads();
    {
      int r = tid >> 4;                 // 16 threads per row
      int c = (tid & 15) * 8;          // 8 contiguous floats each
      int rr = node0 + r; if (rr >= N) rr = N - 1;   // clamp (guarded store below)
      const float* src = agg + ((size_t)rr * NUM_H + h) * DIM + c;
      float4 v0 = *(const float4*)(src);
      float4 v1 = *(const float4*)(src + 4);
      *(float4*)(&tile[r * 132 + c])     = v0;
      *(float4*)(&tile[r * 132 + c + 4]) = v1;
    }
    __syncthreads();
    const float* wb = W + (size_t)h * DIM * DIM + (size_t)(c0 + lh) * DIM + khi;
#pragma unroll 8
    for (int k = 0; k < DIM; k += 4) {
      float2 av = *(const float2*)(&tile[lh * 132 + k + khi]);
      float2 bv = *(const float2*)(wb + k);
      v2f a, b;
      a[0] = av.x; a[1] = av.y;
      b[0] = bv.x; b[1] = bv.y;
      acc = __builtin_amdgcn_wmma_f32_16x16x4_f32(
          /*neg_a=*/false, a, /*neg_b=*/false, b,
          /*c_mod=*/(short)0, acc, /*reuse_a=*/false, /*reuse_b=*/false);
    }
  }

  const int rbase = (lane >> 4) * 8;
#pragma unroll
  for (int vr = 0; vr < 8; ++vr) {
    int row = node0 + rbase + vr;
    if (row < N) out[(size_t)row * DIM + c0 + lh] = acc[vr] * 0.25f;
  }
}

// ---------------------------------------------------------------------------
extern "C" void kernel_launch(void* const* d_in, const int* in_sizes, int n_in,
                              void* d_out, int out_size, void* d_ws, size_t ws_size,
                              hipStream_t stream) {
  const float* msg   = (const float*)d_in[0];   // [E, 128] f32
  const int*   index = (const int*)d_in[1];     // [E] i32
  // d_in[2] = t (unused), d_in[3] = dim_size scalar (derive N from out_size)
  const float* A     = (const float*)d_in[4];   // [4, 128] f32
  const float* W     = (const float*)d_in[5];   // [4, 128, 128] f32
  float* out = (float*)d_out;

  const int E = in_sizes[0] / DIM;
  const int N = out_size / DIM;

  // workspace layout (bytes):
  //   agg    : N*H*D*4    (102.4 MB)  -- zeroed
  //   maxord : N*H*4      (0.8 MB)    -- zeroed (0 < ord(-inf))
  //   denom  : N*H*4      (0.8 MB)    -- zeroed
  //   scores : E*H*4      (16 MB)     -- fully overwritten
  char* ws = (char*)d_ws;
  size_t agg_bytes = (size_t)N * NUM_H * DIM * 4;
  size_t nh_bytes  = (size_t)N * NUM_H * 4;
  float*    agg    = (float*)ws;
  unsigned* maxord = (unsigned*)(ws + agg_bytes);
  float*    denom  = (float*)(ws + agg_bytes + nh_bytes);
  float*    scores = (float*)(ws + agg_bytes + 2 * nh_bytes);

  long zero_dw = (long)((agg_bytes + 2 * nh_bytes) / 4);
  long zero_n4 = (zero_dw + 3) / 4;
  k_zero<<<2048, 256, 0, stream>>>((uint4*)ws, zero_n4);

  k_scores<<<1024, 256, 0, stream>>>(msg, A, scores, E);

  long EH = (long)E * NUM_H;
  int  blkEH = (int)((EH + 255) / 256);
  k_segmax<<<blkEH, 256, 0, stream>>>(scores, index, maxord, EH);
  k_expsum<<<blkEH, 256, 0, stream>>>(scores, index, maxord, denom, EH);

  k_agg<<<2048, 256, 0, stream>>>(msg, index, scores, denom, agg, E);

  int ntiles = (N + 15) / 16;
  k_out<<<ntiles, 256, 0, stream>>>(agg, W, out, N);
}